// Earth4D_80221399154779
// MI455X (gfx1250) — compile-verified
//
#include <hip/hip_runtime.h>

#define NLEVELS 24
#define TBL (1u << 19)          // T
#define PROBE_N 32u             // P
#define CBN 512u                // CB
#define BASE_MASK 16383u        // T/P - 1
#define CB_MASK 511u

#if defined(__has_builtin)
#if __has_builtin(__builtin_amdgcn_global_load_async_to_lds_b32)
#define HAVE_ASYNC_LDS 1
#endif
#if __has_builtin(__builtin_amdgcn_s_wait_asynccnt)
#define HAVE_WAIT_ASYNC 1
#endif
#endif

typedef __attribute__((address_space(1))) int GIPtr;   // global int*
typedef __attribute__((address_space(3))) int LIPtr;   // LDS int*

// ---------------- Pass 3: hash-grid gather, 1 thread = (point,grid,level) ---
// (placed first in the file so the disasm snippet shows its prologue, incl.
//  the async-to-LDS fill of the probe table)
__global__ __launch_bounds__(512) void earth4d_main(
    const float4* __restrict__ c4,
    const unsigned* __restrict__ probe_choice,   // [4*512]
    const float* __restrict__ t0, const float* __restrict__ t1,
    const float* __restrict__ t2, const float* __restrict__ t3,
    float* __restrict__ out, int npts)
{
    __shared__ unsigned sProbe[2048];   // 8 KB: all four grids' probe choices
#ifdef HAVE_ASYNC_LDS
    for (int k = threadIdx.x; k < 2048; k += blockDim.x) {
        __builtin_amdgcn_global_load_async_to_lds_b32(
            (GIPtr*)(probe_choice + k), (LIPtr*)&sProbe[k], 0, 0);
    }
#ifdef HAVE_WAIT_ASYNC
    __builtin_amdgcn_s_wait_asynccnt(0);
#else
    asm volatile("s_wait_asynccnt 0" ::: "memory");
#endif
#else
    for (int k = threadIdx.x; k < 2048; k += blockDim.x)
        sProbe[k] = probe_choice[k];
#endif
    __syncthreads();

    int gid = blockIdx.x * blockDim.x + threadIdx.x;     // < 25.2M, fits int32
    int total = npts * 96;
    if (gid >= total) return;
    int point = gid / 96;
    int sub   = gid - point * 96;        // sub = g*24 + level
    int g     = sub / 24;
    int level = sub - g * 24;

    float4 c = c4[point];
    // grid dim selection: g0=(x,y,z) g1=(x,y,t) g2=(y,z,t) g3=(x,z,t)
    float a = (g == 2) ? c.y : c.x;
    float b = (g <  2) ? c.y : c.z;
    float d = (g == 0) ? c.z : c.w;

    float res = (float)(32u << level);   // exact: 32 * 2^level, level <= 23

    float xa = fminf(fmaxf((a + 1.0f) * 0.5f, 0.0f), 1.0f);
    float xb = fminf(fmaxf((b + 1.0f) * 0.5f, 0.0f), 1.0f);
    float xd = fminf(fmaxf((d + 1.0f) * 0.5f, 0.0f), 1.0f);

    float pa = xa * res, pb = xb * res, pd = xd * res;
    float fa = floorf(pa), fb = floorf(pb), fd = floorf(pd);
    float wa = pa - fa,   wb = pb - fb,   wd = pd - fd;
    unsigned ia = (unsigned)fa, ib = (unsigned)fb, idd = (unsigned)fd;

    const float* tab = (g == 0) ? t0 : (g == 1) ? t1 : (g == 2) ? t2 : t3;
    tab += (size_t)level * (size_t)(TBL * 2u);           // level slice, F=2
    const unsigned* pc = &sProbe[g << 9];

    float acc0 = 0.0f, acc1 = 0.0f;
    #pragma unroll
    for (int ci = 0; ci < 8; ++ci) {
        unsigned bx = (unsigned)(ci & 1);
        unsigned by = (unsigned)((ci >> 1) & 1);
        unsigned bz = (unsigned)((ci >> 2) & 1);
        unsigned cx = ia + bx, cy = ib + by, cz = idd + bz;
        float wt = (bx ? wa : 1.0f - wa) *
                   (by ? wb : 1.0f - wb) *
                   (bz ? wd : 1.0f - wd);
        unsigned h  = cx ^ (cy * 2654435761u) ^ (cz * 805459861u);
        unsigned h2 = (cx * 2165219737u) ^ (cy * 2654435761u) ^ (cz * 3674653429u);
        unsigned idx = ((h & BASE_MASK) << 5) + pc[h2 & CB_MASK];   // ds_load
        const float2 f = *(const float2*)(tab + (size_t)idx * 2u);  // global_load_b64
        acc0 += wt * f.x;
        acc1 += wt * f.y;
    }

    // out[point, g*48 + level*2 + {0,1}] ; streaming (NT) store: 192 MB output
    float* op = out + (size_t)point * 192 + (size_t)sub * 2;
    __builtin_nontemporal_store(acc0, op);
    __builtin_nontemporal_store(acc1, op + 1);
}

// ---------------- Pass 1: per-point ECEF normalize (once, not 96x) ----------
__global__ __launch_bounds__(256) void prep_kernel(const float* __restrict__ coords,
                                                   float4* __restrict__ c4, int n) {
    int i = blockIdx.x * blockDim.x + threadIdx.x;
    if (i >= n) return;
    float lat_deg = coords[i * 4 + 0];
    float lon_deg = coords[i * 4 + 1];
    float elev    = coords[i * 4 + 2];
    float t       = coords[i * 4 + 3];
    const float DEG = 0.017453292519943295f;
    float lat = lat_deg * DEG, lon = lon_deg * DEG;
    float sl = sinf(lat), cl = cosf(lat);
    const float Aa = 6378137.0f, E2 = 0.00669437999014f;
    float Nn = Aa / sqrtf(1.0f - E2 * sl * sl);
    float x = (Nn + elev) * cl * cosf(lon);
    float y = (Nn + elev) * cl * sinf(lon);
    float z = (Nn * (1.0f - E2) + elev) * sl;
    x = x / 6378137.0f;
    y = y / 6378137.0f;
    z = z / 6378137.0f;
    float ts = (t * 2.0f - 1.0f) * 0.9f;
    c4[i] = make_float4(x, y, z, ts);
}

// ---------------- Pass 2: argmax of probe logits -> probe choice ------------
__global__ __launch_bounds__(256) void argmax_kernel(const float* __restrict__ p0,
                                                     const float* __restrict__ p1,
                                                     const float* __restrict__ p2,
                                                     const float* __restrict__ p3,
                                                     unsigned* __restrict__ out) {
    int tid = blockIdx.x * blockDim.x + threadIdx.x;   // 0 .. 2047
    if (tid >= 4 * 512) return;
    int g = tid >> 9;
    int c = tid & 511;
    const float* p = (g == 0) ? p0 : (g == 1) ? p1 : (g == 2) ? p2 : p3;
    p += c * 32;
    float best = p[0];
    unsigned bi = 0;
    #pragma unroll
    for (int j = 1; j < 32; ++j) {
        float v = p[j];
        if (v > best) { best = v; bi = (unsigned)j; }   // first max, like jnp.argmax
    }
    out[tid] = bi;
}

extern "C" void kernel_launch(void* const* d_in, const int* in_sizes, int n_in,
                              void* d_out, int out_size, void* d_ws, size_t ws_size,
                              hipStream_t stream) {
    const float* coords = (const float*)d_in[0];
    const float* xyz_t  = (const float*)d_in[1];
    const float* xyz_p  = (const float*)d_in[2];
    const float* xyt_t  = (const float*)d_in[3];
    const float* xyt_p  = (const float*)d_in[4];
    const float* yzt_t  = (const float*)d_in[5];
    const float* yzt_p  = (const float*)d_in[6];
    const float* xzt_t  = (const float*)d_in[7];
    const float* xzt_p  = (const float*)d_in[8];

    int n = in_sizes[0] / 4;

    float4*   c4 = (float4*)d_ws;                                   // n*16 B
    unsigned* pc = (unsigned*)((char*)d_ws + (size_t)n * 16);       // 8 KB

    prep_kernel<<<(n + 255) / 256, 256, 0, stream>>>(coords, c4, n);
    argmax_kernel<<<8, 256, 0, stream>>>(xyz_p, xyt_p, yzt_p, xzt_p, pc);

    long total = (long)n * 96;
    int blocks = (int)((total + 511) / 512);
    earth4d_main<<<blocks, 512, 0, stream>>>(c4, pc, xyz_t, xyt_t, yzt_t, xzt_t,
                                             (float*)d_out, n);
}